// HdcLevelEncoder_62225486185031
// MI455X (gfx1250) — compile-verified
//
#include <hip/hip_runtime.h>
#include <hip/hip_bf16.h>

#define NSAMP   4096
#define DDIM    10000
#define NLEVELS 1024
#define NCHUNK  16
#define CHUNKN  (NSAMP / NCHUNK)

typedef __attribute__((ext_vector_type(2))) float v2f;
typedef __attribute__((ext_vector_type(8))) float v8f;

// ---------------------------------------------------------------------------
// Kernel 1: 4096-point DFT of the 3 xyz columns via V_WMMA_F32_16X16X4_F32.
//   fr[k,c] = sum_n x[n,c] * cos(-2*pi*k*n/4096)
//   fi[k,c] = sum_n x[n,c] * sin(-2*pi*k*n/4096)
// Each wave computes one 16-frequency tile. Per step of K=4:
//   A (16x4 f32): lane L holds M = L&15, K = 2*(L>>4)+v in VGPR v (cos / sin)
//   B (4x16 f32): lane L holds N = L&15, K = 2*(L>>4)+v in VGPR v
//   C/D (16x16 f32): VGPR r, lane L -> M = r + 8*(L>>4), N = L&15
// Lanes with column >= 3 load a clamped (valid) address instead of masking:
// their garbage only lands in D columns 3..15, which are never read. This
// keeps the inner loop free of EXEC save/restore around the ds_loads.
// Phase computed exactly as (k*n) & 4095 in integer, so no accumulation drift.
// ---------------------------------------------------------------------------
__global__ void __launch_bounds__(256) dft_wmma_kernel(const float* __restrict__ in,
                                                       float* __restrict__ fr,
                                                       float* __restrict__ fi) {
    __shared__ float s_xyz[NSAMP * 3];   // 48 KB of the 320 KB WGP LDS
    const int tid = threadIdx.x;
    for (int i = tid; i < NSAMP * 3; i += 256) {
        int n = i / 3, c = i - n * 3;
        s_xyz[i] = in[n * 4 + 1 + c];
    }
    __syncthreads();

    const int lane  = tid & 31;
    const int ktile = blockIdx.x * 8 + (tid >> 5);   // 32 blocks * 8 waves = 256 tiles
    const int k0    = ktile * 16;

    const int mrow     = lane & 15;      // A row (k offset) / B column
    const int half     = lane >> 4;      // selects K pair {0,1} vs {2,3}
    const int k        = k0 + mrow;
    const int colClamp = (mrow < 3) ? mrow : 2;   // valid address for all lanes

    v8f acc_re = {0.f, 0.f, 0.f, 0.f, 0.f, 0.f, 0.f, 0.f};
    v8f acc_im = {0.f, 0.f, 0.f, 0.f, 0.f, 0.f, 0.f, 0.f};

    const float kNeg2PiOverN = -1.5339807878856412e-3f;  // -2*pi/4096

    for (int n0 = 0; n0 < NSAMP; n0 += 4) {
        const int na0 = n0 + 2 * half;     // sample index for K = 2*half + 0
        const int na1 = na0 + 1;           // sample index for K = 2*half + 1

        const int m0 = (k * na0) & 4095;   // exact phase index
        const int m1 = (k * na1) & 4095;
        float s0, c0, s1, c1;
        __sincosf((float)m0 * kNeg2PiOverN, &s0, &c0);
        __sincosf((float)m1 * kNeg2PiOverN, &s1, &c1);
        v2f a_cos = {c0, c1};
        v2f a_sin = {s0, s1};

        // Unconditional LDS loads (clamped column for lanes >= 3).
        v2f bv = {s_xyz[na0 * 3 + colClamp], s_xyz[na1 * 3 + colClamp]};

        acc_re = __builtin_amdgcn_wmma_f32_16x16x4_f32(false, a_cos, false, bv,
                                                       (short)0, acc_re, false, false);
        acc_im = __builtin_amdgcn_wmma_f32_16x16x4_f32(false, a_sin, false, bv,
                                                       (short)0, acc_im, false, false);
    }

    // Store the 3 useful columns of the 16x16 tile.
    const int krow_base = k0 + half * 8;
    if (mrow < 3) {
        #pragma unroll
        for (int r = 0; r < 8; ++r) {
            fr[(krow_base + r) * 3 + mrow] = acc_re[r];
            fi[(krow_base + r) * 3 + mrow] = acc_im[r];
        }
    }
}

// ---------------------------------------------------------------------------
// Kernel 2: per-sample features -> 16 pre-multiplied row offsets (idx * DDIM).
// jnp.round is round-half-to-even => rintf (v_rndne_f32).
// Offset order: x,y,z,mag,en, xf,yf,zf, xfi,yfi,zfi, magf,magfi, enf,enfi, t
// ---------------------------------------------------------------------------
__device__ __forceinline__ int lvl_off(float v) {
    v = fminf(fmaxf(v, 0.0f), 1.0f);
    return (int)rintf(v * 1023.0f) * DDIM;
}

__global__ void __launch_bounds__(256) index_kernel(const float* __restrict__ in,
                                                    const float* __restrict__ fr,
                                                    const float* __restrict__ fi,
                                                    int* __restrict__ off) {
    const int n = blockIdx.x * 256 + threadIdx.x;
    if (n >= NSAMP) return;

    const float t  = in[n * 4 + 0] - in[0];
    const float x  = in[n * 4 + 1];
    const float y  = in[n * 4 + 2];
    const float z  = in[n * 4 + 3];

    const float ss     = x * x + y * y + z * z;
    const float mags   = sqrtf(ss);
    const float energy = ss * (1.0f / NSAMP);

    const float ar = fr[n * 3 + 0], br = fr[n * 3 + 1], cr = fr[n * 3 + 2];
    const float ai = fi[n * 3 + 0], bi = fi[n * 3 + 1], ci = fi[n * 3 + 2];
    const float sr = ar * ar + br * br + cr * cr;
    const float si = ai * ai + bi * bi + ci * ci;
    const float mfr = sqrtf(sr), mfi = sqrtf(si);
    const float efr = sr * (1.0f / NSAMP), efi = si * (1.0f / NSAMP);

    int* o = off + n * 16;
    o[0]  = lvl_off(x);
    o[1]  = lvl_off(y);
    o[2]  = lvl_off(z);
    o[3]  = lvl_off(mags);
    o[4]  = lvl_off(energy);
    o[5]  = lvl_off(ar);
    o[6]  = lvl_off(br);
    o[7]  = lvl_off(cr);
    o[8]  = lvl_off(ai);
    o[9]  = lvl_off(bi);
    o[10] = lvl_off(ci);
    o[11] = lvl_off(mfr);
    o[12] = lvl_off(mfi);
    o[13] = lvl_off(efr);
    o[14] = lvl_off(efi);
    const float tc = fminf(fmaxf(t, 0.0f), 4096.0f);
    o[15] = (int)rintf(tc * (4095.0f / 4096.0f)) * DDIM;
}

// ---------------------------------------------------------------------------
// Kernel 3: bandwidth-bound gather-accumulate. blockIdx.x tiles D (coalesced
// 128B per wave per gather), blockIdx.y chunks the sample range. Row offsets
// are wave-uniform and loaded as int4 so they become s_load_b128 (SMEM path),
// keeping the vector pipe exclusively for the table gathers.
// ---------------------------------------------------------------------------
__global__ void __launch_bounds__(256) gather_kernel(
    const int* __restrict__ off,
    const float* __restrict__ tx,  const float* __restrict__ ty,
    const float* __restrict__ tz,  const float* __restrict__ tm,
    const float* __restrict__ te,
    const float* __restrict__ txf, const float* __restrict__ tyf,
    const float* __restrict__ tzf,
    const float* __restrict__ txfi, const float* __restrict__ tyfi,
    const float* __restrict__ tzfi,
    const float* __restrict__ tmf,  const float* __restrict__ tmfi,
    const float* __restrict__ tef,  const float* __restrict__ tefi,
    const float* __restrict__ tt,
    float* __restrict__ part) {
    const int d = blockIdx.x * 256 + threadIdx.x;
    if (d >= DDIM) return;
    const int chunk = blockIdx.y;

    float acc = 0.0f;
    const int4* o4 = (const int4*)(off + chunk * CHUNKN * 16);
    for (int i = 0; i < CHUNKN; ++i, o4 += 4) {
        const int4 oa = o4[0];
        const int4 ob = o4[1];
        const int4 oc = o4[2];
        const int4 od = o4[3];
        const float xl  = tx  [oa.x + d];
        const float yl  = ty  [oa.y + d];
        const float zl  = tz  [oa.z + d];
        const float ml  = tm  [oa.w + d];
        const float el  = te  [ob.x + d];
        const float xf  = txf [ob.y + d];
        const float yf  = tyf [ob.z + d];
        const float zf  = tzf [ob.w + d];
        const float xfi = txfi[oc.x + d];
        const float yfi = tyfi[oc.y + d];
        const float zfi = tzfi[oc.z + d];
        const float mf  = tmf [oc.w + d];
        const float mfl = tmfi[od.x + d];
        const float ef  = tef [od.y + d];
        const float efl = tefi[od.z + d];
        const float tl  = tt  [od.w + d];
        acc += (xl * yl * zl + ml + el +
                (xf * yf * zf * xfi * yfi * zfi + mf * mfl + ef * efl)) * tl;
    }
    part[chunk * DDIM + d] = acc;
}

// ---------------------------------------------------------------------------
// Kernel 4: reduce the NCHUNK partials and apply tanh.
// ---------------------------------------------------------------------------
__global__ void __launch_bounds__(256) reduce_tanh_kernel(const float* __restrict__ part,
                                                          float* __restrict__ out) {
    const int d = blockIdx.x * 256 + threadIdx.x;
    if (d >= DDIM) return;
    float s = 0.0f;
    #pragma unroll
    for (int c = 0; c < NCHUNK; ++c) s += part[c * DDIM + d];
    out[d] = tanhf(s);
}

extern "C" void kernel_launch(void* const* d_in, const int* in_sizes, int n_in,
                              void* d_out, int out_size, void* d_ws, size_t ws_size,
                              hipStream_t stream) {
    const float* in   = (const float*)d_in[0];
    const float* tt   = (const float*)d_in[1];
    const float* tx   = (const float*)d_in[2];
    const float* ty   = (const float*)d_in[3];
    const float* tz   = (const float*)d_in[4];
    const float* tm   = (const float*)d_in[5];
    const float* te   = (const float*)d_in[6];
    const float* txf  = (const float*)d_in[7];
    const float* tyf  = (const float*)d_in[8];
    const float* tzf  = (const float*)d_in[9];
    const float* tmf  = (const float*)d_in[10];
    const float* tef  = (const float*)d_in[11];
    const float* txfi = (const float*)d_in[12];
    const float* tyfi = (const float*)d_in[13];
    const float* tzfi = (const float*)d_in[14];
    const float* tmfi = (const float*)d_in[15];
    const float* tefi = (const float*)d_in[16];

    // Workspace carve-up (~1.0 MB total, fully rewritten every call).
    float* fr   = (float*)d_ws;              // 4096*3 f32
    float* fi   = fr + NSAMP * 3;            // 4096*3 f32
    int*   off  = (int*)(fi + NSAMP * 3);    // 4096*16 i32 (pre-multiplied row offsets)
    float* part = (float*)(off + NSAMP * 16);// NCHUNK*10000 f32

    dft_wmma_kernel<<<dim3(32), dim3(256), 0, stream>>>(in, fr, fi);
    index_kernel<<<dim3(NSAMP / 256), dim3(256), 0, stream>>>(in, fr, fi, off);
    gather_kernel<<<dim3((DDIM + 255) / 256, NCHUNK), dim3(256), 0, stream>>>(
        off, tx, ty, tz, tm, te, txf, tyf, tzf, txfi, tyfi, tzfi, tmf, tmfi, tef, tefi,
        tt, part);
    reduce_tanh_kernel<<<dim3((DDIM + 255) / 256), dim3(256), 0, stream>>>(
        part, (float*)d_out);
}